// TemporalGNN_55731495633400
// MI455X (gfx1250) — compile-verified
//
#include <hip/hip_runtime.h>
#include <math.h>

#define NNODE 100000
#define FDIM  128
#define NEDGE 625000

typedef __attribute__((ext_vector_type(2))) float v2f;
typedef __attribute__((ext_vector_type(8))) float v8f;

__device__ __forceinline__ float wave_sum(float v) {
#pragma unroll
  for (int off = 16; off > 0; off >>= 1) v += __shfl_xor(v, off, 32);
  return v;
}

// ---- CDNA5 async global->LDS copy (ASYNCcnt path) ----
__device__ __forceinline__ void async_g2l_b32(unsigned lds_addr, const float* g) {
  asm volatile("global_load_async_to_lds_b32 %0, %1, off"
               :: "v"(lds_addr), "v"(g) : "memory");
}
__device__ __forceinline__ void async_wait0() {
  asm volatile("s_wait_asynccnt 0" ::: "memory");
}
__device__ __forceinline__ unsigned lds_off(const void* p) {
  return (unsigned)(uintptr_t)p;   // low 32 bits of a generic LDS pointer = LDS byte offset
}

// ---------------- utility kernels ----------------

__global__ void k_fill(float* __restrict__ p, float val, int n) {
  int i = blockIdx.x * blockDim.x + threadIdx.x;
  int stride = gridDim.x * blockDim.x;
  for (; i < n; i += stride) p[i] = val;
}

__global__ void k_relu(float* __restrict__ p, int n) {
  int i = blockIdx.x * blockDim.x + threadIdx.x;
  int stride = gridDim.x * blockDim.x;
  for (; i < n; i += stride) { float v = p[i]; p[i] = v > 0.f ? v : 0.f; }
}

__global__ void k_deg_accum(const int* __restrict__ ei, const float* __restrict__ ew,
                            float* __restrict__ deg) {
  int e = blockIdx.x * blockDim.x + threadIdx.x;
  if (e < NEDGE) atomicAdd(&deg[ei[NEDGE + e]], ew[e]);
}

__global__ void k_deg_finish(float* __restrict__ dinv) {
  int i = blockIdx.x * blockDim.x + threadIdx.x;
  if (i < NNODE) { float v = dinv[i]; dinv[i] = v > 0.f ? rsqrtf(v) : 0.f; }
}

__global__ void k_rnorm(const float* __restrict__ p, float* __restrict__ rnorm) {
  int lane = threadIdx.x;
  float s = 0.f;
#pragma unroll
  for (int i = 0; i < 4; ++i) { float v = p[lane + 32 * i]; s += v * v; }
  s = wave_sum(s);
  if (lane == 0) rnorm[0] = rsqrtf(s);
}

__global__ void k_score(const float* __restrict__ x, const float* __restrict__ p,
                        const float* __restrict__ rnorm, float* __restrict__ score) {
  int wave = threadIdx.x >> 5, lane = threadIdx.x & 31;
  int row = blockIdx.x * (blockDim.x >> 5) + wave;
  if (row >= NNODE) return;
  const float* xr = x + (size_t)row * FDIM;
  float s = 0.f;
#pragma unroll
  for (int i = 0; i < 4; ++i) s += xr[lane + 32 * i] * p[lane + 32 * i];
  s = wave_sum(s);
  if (lane == 0) score[row] = s * rnorm[0];
}

// iterative top-128 arg-max (single block; ties -> lowest index, like lax.top_k)
__global__ void k_topk(float* __restrict__ score, int* __restrict__ tidx,
                       float* __restrict__ tvals) {
  __shared__ float sv[1024];
  __shared__ int   si[1024];
  int tid = threadIdx.x;
  for (int it = 0; it < FDIM; ++it) {
    float best = -INFINITY; int bidx = 0x7fffffff;
    for (int i = tid; i < NNODE; i += 1024) {
      float v = score[i];
      if (v > best || (v == best && i < bidx)) { best = v; bidx = i; }
    }
    sv[tid] = best; si[tid] = bidx;
    __syncthreads();
    for (int s = 512; s > 0; s >>= 1) {
      if (tid < s) {
        float v2 = sv[tid + s]; int i2 = si[tid + s];
        if (v2 > sv[tid] || (v2 == sv[tid] && i2 < si[tid])) { sv[tid] = v2; si[tid] = i2; }
      }
      __syncthreads();
    }
    if (tid == 0) {
      tidx[it] = si[0]; tvals[it] = sv[0];
      score[si[0]] = -INFINITY;
    }
    __syncthreads();
  }
}

__global__ void k_gather(const float* __restrict__ x, const int* __restrict__ tidx,
                         const float* __restrict__ tvals, float* __restrict__ xt) {
  int r = blockIdx.x, j = threadIdx.x;
  float g = tanhf(tvals[r]);
  xt[r * FDIM + j] = x[(size_t)tidx[r] * FDIM + j] * g;
}

// ---------------- WMMA f32 16x16x4 GEMMs ----------------
// A-frag (16x4): lanes 0-15 -> K={0,1}, lanes 16-31 -> K={2,3}; M = lane&15.
// B-frag (4x16): same K split; N = lane&15.
// C/D: 8 VGPRs, M = r + 8*(lane>>4), N = lane&15.

// Big GEMM specialized: C[M,128] = A[M,128] * B[128,128], M % 16 == 0.
// A tile + weight staged into LDS via async global->LDS; inner loop = ds_load + wmma.
#define APAD 132   // 16-row A tile, padded stride (banks 4m -> conflict free)
#define BPAD 66    // 64-K weight chunk, transposed [n][k], padded (banks 2n)
__global__ void k_gemm_nn_lds(const float* __restrict__ A, const float* __restrict__ B,
                              float* __restrict__ C, int M) {
  __shared__ float sA[16 * APAD];      // 8.25 KB
  __shared__ float sB[128 * BPAD];     // 33   KB (one 64-wide K chunk, transposed)
  int tid  = threadIdx.x;
  int lane = tid & 31;
  int wave = tid >> 5;                 // ntile = wave (8 waves cover 128 cols)
  int mtile = blockIdx.x;
  int l15 = lane & 15;
  int kb  = (lane >> 4) << 1;          // 0 or 2
  unsigned lA = lds_off(sA);
  unsigned lB = lds_off(sB);

  // stage the 16x128 A tile (contiguous 8KB in global) with row padding
  const float* gA = A + (size_t)mtile * 16 * FDIM;
#pragma unroll
  for (int i = 0; i < 8; ++i) {
    int idx = tid + 256 * i;                       // 2048 elements
    int r = idx >> 7, c = idx & 127;
    async_g2l_b32(lA + (unsigned)(r * APAD + c) * 4u, gA + idx);
  }

  v8f acc = {};
#pragma unroll
  for (int kc = 0; kc < 2; ++kc) {
    if (kc > 0) __syncthreads();                   // everyone done reading previous chunk
    // stage weight rows k = kc*64 .. kc*64+63, transposed into sB[n*BPAD + k']
    const float* gB = B + kc * 64 * FDIM;
#pragma unroll
    for (int i = 0; i < 32; ++i) {
      int idx = tid + 256 * i;                     // 8192 elements
      int k = idx >> 7, n = idx & 127;
      async_g2l_b32(lB + (unsigned)(n * BPAD + k) * 4u, gB + idx);
    }
    async_wait0();
    __syncthreads();

    const float* aP = &sA[l15 * APAD + kc * 64 + kb];
    const float* bP = &sB[(wave * 16 + l15) * BPAD + kb];
#pragma unroll
    for (int k0 = 0; k0 < 64; k0 += 4) {
      v2f a = *(const v2f*)(aP + k0);
      v2f b = *(const v2f*)(bP + k0);
      acc = __builtin_amdgcn_wmma_f32_16x16x4_f32(false, a, false, b, (short)0, acc,
                                                  false, false);
    }
  }

  float* crow = C + (size_t)(mtile * 16 + 8 * (lane >> 4)) * FDIM + wave * 16 + l15;
#pragma unroll
  for (int r = 0; r < 8; ++r) crow[(size_t)r * FDIM] = acc[r];
}

// small GRU GEMMs: C[M,Nn] = A[M,K] * B[Nn,K]^T  (B row-major [Nn,K])
__global__ void k_gemm_nt(const float* __restrict__ A, const float* __restrict__ B,
                          float* __restrict__ C, int M, int Nn, int K) {
  int lane = threadIdx.x & 31;
  int wave = threadIdx.x >> 5;
  int mtile = blockIdx.x;
  int ntile = blockIdx.y * (blockDim.x >> 5) + wave;
  if (mtile * 16 >= M || ntile * 16 >= Nn) return;   // whole-wave exit, EXEC stays full
  int l15 = lane & 15;
  int kb  = (lane >> 4) << 1;
  const float* arow = A + (size_t)(mtile * 16 + l15) * K + kb;
  const float* brow = B + (size_t)(ntile * 16 + l15) * K + kb;
  v8f acc = {};
#pragma unroll 4
  for (int k0 = 0; k0 < K; k0 += 4) {
    v2f a = *(const v2f*)(arow + k0);
    v2f b = *(const v2f*)(brow + k0);
    acc = __builtin_amdgcn_wmma_f32_16x16x4_f32(false, a, false, b, (short)0, acc,
                                                false, false);
  }
  float* crow = C + (size_t)(mtile * 16 + 8 * (lane >> 4)) * Nn + ntile * 16 + l15;
#pragma unroll
  for (int r = 0; r < 8; ++r) crow[(size_t)r * Nn] = acc[r];
}

// GRU gate fusion: Wn = (1-z)*n + z*h0
__global__ void k_gru_combine(const float* __restrict__ gi, const float* __restrict__ gh,
                              const float* __restrict__ bih, const float* __restrict__ bhh,
                              const float* __restrict__ h0, float* __restrict__ Wn) {
  int i = blockIdx.x, j = threadIdx.x;
  float ir  = gi[i * 384 + j]       + bih[j];
  float iz  = gi[i * 384 + 128 + j] + bih[128 + j];
  float inn = gi[i * 384 + 256 + j] + bih[256 + j];
  float hr  = gh[i * 384 + j]       + bhh[j];
  float hz  = gh[i * 384 + 128 + j] + bhh[128 + j];
  float hn  = gh[i * 384 + 256 + j] + bhh[256 + j];
  float r = 1.f / (1.f + expf(-(ir + hr)));
  float z = 1.f / (1.f + expf(-(iz + hz)));
  float n = tanhf(inn + r * hn);
  Wn[i * 128 + j] = (1.f - z) * n + z * h0[i * 128 + j];
}

// GCN scatter: out[col,:] += dinv[row]*w*dinv[col] * xw[row,:]  (1 wave per edge)
__global__ void k_scatter(const int* __restrict__ ei, const float* __restrict__ ew,
                          const float* __restrict__ dinv, const float* __restrict__ xw,
                          float* __restrict__ out) {
  int e = blockIdx.x * (blockDim.x >> 5) + (threadIdx.x >> 5);
  if (e >= NEDGE + NNODE) return;
  int lane = threadIdx.x & 31;
  int row, col; float w;
  if (e < NEDGE) { row = ei[e]; col = ei[NEDGE + e]; w = ew[e]; }
  else           { row = col = e - NEDGE; w = 1.0f; }              // self loop
  float nrm = dinv[row] * w * dinv[col];
  const float* src = xw + (size_t)row * FDIM;
  float* dst = out + (size_t)col * FDIM;
#pragma unroll
  for (int i = 0; i < 4; ++i)
    atomicAdd(&dst[lane + 32 * i], nrm * src[lane + 32 * i]);
}

// fused head
__global__ void k_final(const float* __restrict__ h, const float* __restrict__ w1,
                        const float* __restrict__ b1, const float* __restrict__ w2,
                        const float* __restrict__ b2, float* __restrict__ out) {
  int wave = threadIdx.x >> 5, lane = threadIdx.x & 31;
  int row = blockIdx.x * (blockDim.x >> 5) + wave;
  if (row >= NNODE) return;
  const float* hr = h + (size_t)row * FDIM;
  float a0 = 0.f, a1 = 0.f;
#pragma unroll
  for (int i = 0; i < 4; ++i) {
    float v = hr[lane + 32 * i];
    a0 += v * w1[lane + 32 * i];
    a1 += v * w1[FDIM + lane + 32 * i];
  }
  a0 = wave_sum(a0); a1 = wave_sum(a1);
  if (lane == 0)
    out[row] = b2[0] + w2[0] * (a0 + b1[0]) + w2[1] * (a1 + b1[1]);
}

// ---------------- host launcher ----------------

extern "C" void kernel_launch(void* const* d_in, const int* in_sizes, int n_in,
                              void* d_out, int out_size, void* d_ws, size_t ws_size,
                              hipStream_t stream) {
  (void)in_sizes; (void)n_in; (void)out_size; (void)ws_size;
  const float* x   = (const float*)d_in[0];
  const int*   ei  = (const int*)d_in[1];
  const float* ew  = (const float*)d_in[2];
  const float* p[2]   = {(const float*)d_in[3],  (const float*)d_in[9]};
  const float* wih[2] = {(const float*)d_in[4],  (const float*)d_in[10]};
  const float* whh[2] = {(const float*)d_in[5],  (const float*)d_in[11]};
  const float* bih[2] = {(const float*)d_in[6],  (const float*)d_in[12]};
  const float* bhh[2] = {(const float*)d_in[7],  (const float*)d_in[13]};
  const float* h0[2]  = {(const float*)d_in[8],  (const float*)d_in[14]};
  const float* l1w = (const float*)d_in[15];
  const float* l1b = (const float*)d_in[16];
  const float* l2w = (const float*)d_in[17];
  const float* l2b = (const float*)d_in[18];
  float* out = (float*)d_out;

  float* ws = (float*)d_ws;
  const size_t NF = (size_t)NNODE * FDIM;
  float* bufA  = ws;                   // xw  [N,F]
  float* bufH  = bufA + NF;            // h   [N,F]
  float* score = bufH + NF;            // [N]
  float* dinv  = score + NNODE;        // [N]
  float* xt    = dinv + NNODE;         // [F,F]
  float* Wn    = xt + FDIM * FDIM;     // [F,F]
  float* gi    = Wn + FDIM * FDIM;     // [F,3F]
  float* gh    = gi + FDIM * 3 * FDIM; // [F,3F]
  float* tvals = gh + FDIM * 3 * FDIM; // [F]
  float* rnorm = tvals + FDIM;         // [1] (+pad)
  int*   tidx  = (int*)(rnorm + 8);    // [F]

  // symmetric-normalized degree (shared by both layers; deg starts at 1 = self loop)
  k_fill<<<(NNODE + 255) / 256, 256, 0, stream>>>(dinv, 1.0f, NNODE);
  k_deg_accum<<<(NEDGE + 255) / 256, 256, 0, stream>>>(ei, ew, dinv);
  k_deg_finish<<<(NNODE + 255) / 256, 256, 0, stream>>>(dinv);

  const float* in = x;
  for (int l = 0; l < 2; ++l) {
    k_rnorm<<<1, 32, 0, stream>>>(p[l], rnorm);
    k_score<<<(NNODE + 7) / 8, 256, 0, stream>>>(in, p[l], rnorm, score);
    k_topk<<<1, 1024, 0, stream>>>(score, tidx, tvals);
    k_gather<<<FDIM, FDIM, 0, stream>>>(in, tidx, tvals, xt);
    // GRU gate GEMMs: [128,384] = [128,128] x [384,128]^T
    k_gemm_nt<<<dim3(8, 3), 256, 0, stream>>>(xt,    wih[l], gi, 128, 384, 128);
    k_gemm_nt<<<dim3(8, 3), 256, 0, stream>>>(h0[l], whh[l], gh, 128, 384, 128);
    k_gru_combine<<<FDIM, FDIM, 0, stream>>>(gi, gh, bih[l], bhh[l], h0[l], Wn);
    // big GEMM: xw = in @ Wn, LDS-staged via async global->LDS (100000 = 6250*16)
    k_gemm_nn_lds<<<NNODE / 16, 256, 0, stream>>>(in, Wn, bufA, NNODE);
    // GCN aggregation
    k_fill<<<(int)((NF + 255) / 256), 256, 0, stream>>>(bufH, 0.0f, (int)NF);
    k_scatter<<<(NEDGE + NNODE + 7) / 8, 256, 0, stream>>>(ei, ew, dinv, bufA, bufH);
    k_relu<<<(int)((NF + 255) / 256), 256, 0, stream>>>(bufH, (int)NF);
    in = bufH;
  }
  k_final<<<(NNODE + 7) / 8, 256, 0, stream>>>(bufH, l1w, l1b, l2w, l2b, out);
}